// DotProductScorer_7284264534433
// MI455X (gfx1250) — compile-verified
//
#include <hip/hip_runtime.h>

typedef float v2f __attribute__((ext_vector_type(2)));
typedef float v4f __attribute__((ext_vector_type(4)));
typedef float v8f __attribute__((ext_vector_type(8)));

// ---------------------------------------------------------------------------
// Zero-fill d_out (harness poisons it; tokens outside [starts[0],starts[-1])
// must be 0 per the reference's new_zeros semantics).
// ---------------------------------------------------------------------------
__global__ __launch_bounds__(256) void zero_f32_kernel(float* __restrict__ out, int n) {
    int i = (blockIdx.x * blockDim.x + threadIdx.x) * 4;
    if (i + 3 < n) {
        *(v4f*)(out + i) = (v4f){0.f, 0.f, 0.f, 0.f};
    } else {
        for (int j = i; j < n; ++j) out[j] = 0.f;
    }
}

// ---------------------------------------------------------------------------
// C[M,N] = A[M,Kd] @ W[N,Kd]^T (+ bias[N]), fp32 via V_WMMA_F32_16X16X4_F32.
// One wave per 16x16 C tile.
//
// Fragment layout (ISA 7.12.2, 32-bit, wave32):
//   A 16x4:  lane L -> row m = L&15;  element j holds K = 2*(L>>4) + j
//   B 4x16:  lane L -> col n = L&15;  element j holds K = 2*(L>>4) + j
//   C 16x16: lane L -> col n = L&15;  element r holds row r + 8*(L>>4)
// Since B[k][n] = W[n][k], both A and B fragments are contiguous float2
// loads from row-major A / W.
// ---------------------------------------------------------------------------
__global__ __launch_bounds__(32) void gemm_nt_bias_wmma_kernel(
    const float* __restrict__ A, const float* __restrict__ W,
    const float* __restrict__ bias, float* __restrict__ C,
    int N, int Kd)
{
    const int ntn  = N >> 4;
    const int tm   = blockIdx.x / ntn;
    const int tn   = blockIdx.x % ntn;
    const int lane = threadIdx.x & 31;
    const int half = lane >> 4;   // 0 or 1
    const int mn   = lane & 15;

    const float* ar = A + (size_t)(tm * 16 + mn) * Kd + 2 * half;
    const float* br = W + (size_t)(tn * 16 + mn) * Kd + 2 * half;

    v8f acc = {};
    for (int k = 0; k < Kd; k += 4) {
        v2f a = *(const v2f*)(ar + k);
        v2f b = *(const v2f*)(br + k);
        acc = __builtin_amdgcn_wmma_f32_16x16x4_f32(
            /*neg_a=*/false, a, /*neg_b=*/false, b,
            /*c_mod=*/(short)0, acc, /*reuse_a=*/false, /*reuse_b=*/false);
    }

    const float bv = bias ? bias[tn * 16 + mn] : 0.0f;
    float* cr = C + (size_t)(tm * 16 + half * 8) * N + (tn * 16 + mn);
#pragma unroll
    for (int r = 0; r < 8; ++r)
        cr[(size_t)r * N] = acc[r] + bv;
}

// ---------------------------------------------------------------------------
// Per-token dot product against the segment's projected query.
// One block (8 wave32s) per segment b. Each wave consumes one token per
// iteration: 32 lanes x float4 = 512B = one full token row -> a single
// perfectly-coalesced global_load_b128 per lane, non-temporal (read-once
// 1 GB stream). 5-step shfl_xor wave reduction; lane 0 stores the logit.
// Requires D == 128 (reference: D_TOKEN = 128).
// ---------------------------------------------------------------------------
__global__ __launch_bounds__(256) void segment_dot_kernel(
    const float* __restrict__ cand, const float* __restrict__ kq,
    const int* __restrict__ starts, float* __restrict__ out, int D)
{
    const int b      = blockIdx.x;
    const int lane   = threadIdx.x & 31;
    const int wave   = threadIdx.x >> 5;
    const int nwaves = blockDim.x >> 5;

    // This wave's slice of the segment's projected query (4 floats/lane).
    const v4f kv = *(const v4f*)(kq + (size_t)b * D + lane * 4);

    const int start = starts[b];
    const int end   = starts[b + 1];

    for (int tok = start + wave; tok < end; tok += nwaves) {
        const v4f* p = (const v4f*)(cand + (size_t)tok * D) + lane;
        // prefetch ~4 iterations (16 KB) ahead of this wave's stream
        __builtin_prefetch(cand + (size_t)(tok + nwaves * 4) * D + lane * 4, 0, 0);
        v4f c = __builtin_nontemporal_load(p);
        float s = c.x * kv.x + c.y * kv.y + c.z * kv.z + c.w * kv.w;
        s += __shfl_xor(s, 16, 32);
        s += __shfl_xor(s,  8, 32);
        s += __shfl_xor(s,  4, 32);
        s += __shfl_xor(s,  2, 32);
        s += __shfl_xor(s,  1, 32);
        if (lane == 0) out[tok] = s;
    }
}

// ---------------------------------------------------------------------------
// Launch: zero(out); q = sv@Wq^T + bq; kq = q@Wk^T; score tokens.
// ---------------------------------------------------------------------------
extern "C" void kernel_launch(void* const* d_in, const int* in_sizes, int n_in,
                              void* d_out, int out_size, void* d_ws, size_t ws_size,
                              hipStream_t stream) {
    const float* state_vec = (const float*)d_in[0];   // [B, Ds]
    const float* cand      = (const float*)d_in[1];   // [K, Dt]
    const int*   starts    = (const int*)  d_in[2];   // [B+1]
    const float* Wq        = (const float*)d_in[3];   // [Dt, Ds]
    const float* bq        = (const float*)d_in[4];   // [Dt]
    const float* Wk        = (const float*)d_in[5];   // [Dt, Dt]
    float* out = (float*)d_out;

    const int B  = in_sizes[2] - 1;        // 4096
    const int Dt = in_sizes[4];            // 128
    const int Ds = in_sizes[0] / B;        // 256
    (void)n_in; (void)ws_size;

    float* q_ws  = (float*)d_ws;                   // [B, Dt]
    float* kq_ws = q_ws + (size_t)B * Dt;          // [B, Dt]

    // 1) zero output (covers any tokens outside [starts[0], starts[-1]))
    const int zt = 256;
    const int zb = (out_size + zt * 4 - 1) / (zt * 4);
    zero_f32_kernel<<<zb, zt, 0, stream>>>(out, out_size);

    // 2) q = state_vec @ Wq^T + bq   (fp32 WMMA)
    gemm_nt_bias_wmma_kernel<<<(B / 16) * (Dt / 16), 32, 0, stream>>>(
        state_vec, Wq, bq, q_ws, Dt, Ds);

    // 3) kq = q @ Wk^T               (fp32 WMMA)
    gemm_nt_bias_wmma_kernel<<<(B / 16) * (Dt / 16), 32, 0, stream>>>(
        q_ws, Wk, nullptr, kq_ws, Dt, Dt);

    // 4) streaming per-token dots (HBM-bound: ~1.07 GB @ 23.3 TB/s ~ 46 us)
    segment_dot_kernel<<<B, 256, 0, stream>>>(cand, kq_ws, starts, out, Dt);
}